// APNet_15023795601945
// MI455X (gfx1250) — compile-verified
//
#include <hip/hip_runtime.h>

// ---------------------------------------------------------------------------
// APNet symmetry-function forward for MI455X (gfx1250, wave32).
//   K1: radial SFs  — grid B*A,  64 thr/block
//   K2: angular SFs — grid B*A,  64 thr/block
//        * pairs-mask tile staged via global_load_async_to_lds_b128 (ASYNCcnt)
//        * 60x40 Gram matrix p_j.p_k via v_wmma_f32_16x16x4_f32
//   K3: pair dists  — grid B*NA, 64 thr/block
// ---------------------------------------------------------------------------

namespace {
constexpr int B_    = 8;
constexpr int A_    = 120;
constexpr int NA_   = 60;
constexpr int NB_   = 60;
constexpr int NJ_   = 60;
constexpr int NK_   = 40;
constexpr int NNEI_ = 119;
constexpr int NRAD_ = 43;
constexpr int NAP_  = 21;

constexpr float EPSf  = 1e-12f;
constexpr float PI_8  = 0.39269908169872414f;   // pi/8
constexpr float PI_9  = 0.34906585039886590f;   // pi/9

// flat float offsets into d_out (reference returns tuple, concatenated flat)
constexpr long long OFF_RA  = 0;
constexpr long long OFF_RB  = OFF_RA  + (long long)B_ * NA_ * 5 * NRAD_;        // 103200
constexpr long long OFF_APA = OFF_RB  + (long long)B_ * NB_ * 5 * NRAD_;        // 206400
constexpr long long OFF_APB = OFF_APA + (long long)B_ * NA_ * NJ_ * 5 * NAP_;   // 3230400
constexpr long long OFF_D   = OFF_APB + (long long)B_ * NB_ * NJ_ * 5 * NAP_;   // 6254400
} // namespace

typedef __attribute__((ext_vector_type(2))) float v2f;
typedef __attribute__((ext_vector_type(8))) float v8f;

__device__ __forceinline__ float fcut(float d, float pi_over_rc, float rc) {
    return (d < rc) ? 0.5f * (__cosf(d * pi_over_rc) + 1.0f) : 0.0f;
}

// ============================ K1: radial ====================================
__global__ __launch_bounds__(64) void apnet_radial(
    const float* __restrict__ positions,     // (B,A,3)
    const float* __restrict__ cell,          // (B,3,3)
    const float* __restrict__ cell_offset,   // (B,A,NNEI,3)
    const float* __restrict__ z_weight,      // (10,5)
    const int*   __restrict__ Zat,           // (B,A)
    const int*   __restrict__ neighbors,     // (B,A,NNEI)
    const float* __restrict__ nbr_mask,      // (B,A,NNEI)
    float*       __restrict__ out)
{
    __shared__ float s_d[NNEI_];
    __shared__ float s_c[NNEI_];
    __shared__ float s_z[NNEI_][5];

    const int ba  = blockIdx.x;
    const int b   = ba / A_;
    const int a   = ba % A_;
    const int tid = threadIdx.x;

    const float* cb = cell + b * 9;
    const float c00=cb[0],c01=cb[1],c02=cb[2],c10=cb[3],c11=cb[4],c12=cb[5],c20=cb[6],c21=cb[7],c22=cb[8];
    const float pix = positions[(b*A_+a)*3+0];
    const float piy = positions[(b*A_+a)*3+1];
    const float piz = positions[(b*A_+a)*3+2];

    for (int n = tid; n < NNEI_; n += 64) {
        const int   jj = neighbors[(long long)ba * NNEI_ + n];
        const float* pp = positions + (long long)(b*A_ + jj) * 3;
        const float* of = cell_offset + ((long long)ba * NNEI_ + n) * 3;
        const float dx = pp[0] - pix + of[0]*c00 + of[1]*c10 + of[2]*c20;
        const float dy = pp[1] - piy + of[0]*c01 + of[1]*c11 + of[2]*c21;
        const float dz = pp[2] - piz + of[0]*c02 + of[1]*c12 + of[2]*c22;
        float d = __builtin_sqrtf(fmaxf(dx*dx + dy*dy + dz*dz, EPSf));
        const float m = nbr_mask[(long long)ba * NNEI_ + n];
        d = (m != 0.0f) ? d : 0.0f;
        s_d[n] = d;
        s_c[n] = fcut(d, PI_8, 8.0f) * m;
        const int ze = Zat[b*A_ + jj];
        #pragma unroll
        for (int e = 0; e < 5; ++e) s_z[n][e] = z_weight[ze*5 + e];
    }
    __syncthreads();

    if (tid < NRAD_) {
        const float offr = 0.8f + 0.1f * (float)tid;     // linspace(0.8, 5.0, 43)
        float a0=0.f, a1=0.f, a2=0.f, a3=0.f, a4=0.f;
        for (int n = 0; n < NNEI_; ++n) {
            const float t = s_d[n] - offr;
            const float g = __expf(-100.0f * t * t) * s_c[n];   // coeff = -0.5/(0.5*0.1^2)
            a0 += s_z[n][0]*g; a1 += s_z[n][1]*g; a2 += s_z[n][2]*g;
            a3 += s_z[n][3]*g; a4 += s_z[n][4]*g;
        }
        const long long base = (a < NA_)
            ? OFF_RA + (long long)(b*NA_ + a) * (5*NRAD_)
            : OFF_RB + (long long)(b*NB_ + (a - NA_)) * (5*NRAD_);
        out[base + 0*NRAD_ + tid] = a0;
        out[base + 1*NRAD_ + tid] = a1;
        out[base + 2*NRAD_ + tid] = a2;
        out[base + 3*NRAD_ + tid] = a3;
        out[base + 4*NRAD_ + tid] = a4;
    }
}

// ============================ K2: angular ===================================
__global__ __launch_bounds__(64) void apnet_angular(
    const float* __restrict__ positions,   // (B,A,3)
    const float* __restrict__ cell,        // (B,3,3)
    const float* __restrict__ noi,         // neighbor_offset_inter (B,A,NJ,3)
    const float* __restrict__ coi,         // cell_offset_intra     (B,A,NK,3)
    const float* __restrict__ z_weight,    // (10,5)
    const int*   __restrict__ Zat,         // (B,A)
    const int*   __restrict__ idx_j,       // (B,A,NJ)
    const int*   __restrict__ idx_k,       // (B,A,NK)
    const float* __restrict__ pairs_mask,  // (B,A,NJ,NK)
    float*       __restrict__ out)
{
    __shared__ float4 s_pj4[64];              // padded j positions (x,y,z,0)
    __shared__ float4 s_pk4[48];              // padded k positions
    __shared__ float4 s_jd[64];               // (rij^2, 0.5/rij, |p_j|^2, fcij)
    __shared__ float4 s_kd[48];               // (rik^2, 1.0/rik, |p_k|^2, fcik)
    __shared__ float  s_zk[48][5];
    __shared__ float  s_cos[64][49];          // odd stride: conflict-free cols
    __shared__ __align__(16) float s_pm[NJ_ * NK_];   // staged pairs-mask tile

    const int ba  = blockIdx.x;
    const int b   = ba / A_;
    const int a   = ba % A_;
    const int tid = threadIdx.x;

    // -------- phase 0: async-copy the (60x40) pairs-mask tile into LDS -----
    // global_load_async_to_lds_b128: per-lane 16B, tracked by ASYNCcnt.
    {
        const float* gbase = pairs_mask + (long long)ba * (NJ_ * NK_);
        const unsigned lds_base = (unsigned)(unsigned long long)&s_pm[0];
        for (int i = tid; i < (NJ_ * NK_) / 4; i += 64) {
            const unsigned voff = (unsigned)i * 16u;
            const unsigned ldsa = lds_base + voff;
            asm volatile("global_load_async_to_lds_b128 %0, %1, %2"
                         :
                         : "v"(ldsa), "v"(voff), "s"((unsigned long long)gbase)
                         : "memory");
        }
    }

    const float* cb = cell + b * 9;
    const float c00=cb[0],c01=cb[1],c02=cb[2],c10=cb[3],c11=cb[4],c12=cb[5],c20=cb[6],c21=cb[7],c22=cb[8];
    const float pix = positions[(b*A_+a)*3+0];
    const float piy = positions[(b*A_+a)*3+1];
    const float piz = positions[(b*A_+a)*3+2];

    // -------- phase A: stage j / k geometry + embeddings into LDS ----------
    {
        const int j = tid;                     // 0..63 (rows >= 60 zero-padded)
        float x=0.f, y=0.f, z=0.f, rij=1.f, fc=0.f;
        if (j < NJ_) {
            const int    jj = idx_j[(long long)ba * NJ_ + j];
            const float* pp = positions + (long long)(b*A_ + jj) * 3;
            const float* of = noi + ((long long)ba * NJ_ + j) * 3;
            x = pp[0] + of[0]*c00 + of[1]*c10 + of[2]*c20;
            y = pp[1] + of[0]*c01 + of[1]*c11 + of[2]*c21;
            z = pp[2] + of[0]*c02 + of[1]*c12 + of[2]*c22;
            const float dx=x-pix, dy=y-piy, dz=z-piz;
            rij = __builtin_sqrtf(fmaxf(dx*dx+dy*dy+dz*dz, EPSf));
            fc  = fcut(rij, PI_8, 8.0f);
        }
        s_pj4[j] = make_float4(x, y, z, 0.f);
        s_jd[j]  = make_float4(rij*rij, __fdividef(0.5f, rij), x*x + y*y + z*z, fc);
    }
    if (tid < 48) {
        const int k = tid;                     // 0..47 (rows >= 40 zero-padded)
        float x=0.f, y=0.f, z=0.f, rik=1.f, fc=0.f;
        float zr[5] = {0.f,0.f,0.f,0.f,0.f};
        if (k < NK_) {
            const int    kk = idx_k[(long long)ba * NK_ + k];
            const float* pp = positions + (long long)(b*A_ + kk) * 3;
            const float* of = coi + ((long long)ba * NK_ + k) * 3;
            x = pp[0] + of[0]*c00 + of[1]*c10 + of[2]*c20;
            y = pp[1] + of[0]*c01 + of[1]*c11 + of[2]*c21;
            z = pp[2] + of[0]*c02 + of[1]*c12 + of[2]*c22;
            const float dx=x-pix, dy=y-piy, dz=z-piz;
            rik = __builtin_sqrtf(fmaxf(dx*dx+dy*dy+dz*dz, EPSf));
            fc  = fcut(rik, PI_8, 8.0f);
            const int ze = Zat[b*A_ + kk];
            #pragma unroll
            for (int e = 0; e < 5; ++e) zr[e] = z_weight[ze*5 + e];
        }
        s_pk4[k] = make_float4(x, y, z, 0.f);
        s_kd[k]  = make_float4(rik*rik, __fdividef(1.0f, rik), x*x + y*y + z*z, fc);
        #pragma unroll
        for (int e = 0; e < 5; ++e) s_zk[k][e] = zr[e];
    }

    // async mask tile must have landed before anyone reads s_pm (phase C)
    asm volatile("s_wait_asynccnt 0x0" ::: "memory");
    __syncthreads();

    // -------- phase B: Gram matrix p_j . p_k via v_wmma_f32_16x16x4_f32 ----
    // r_jk^2 = |p_j|^2 + |p_k|^2 - 2 p_j.p_k ; then cos(theta_jk) -> LDS.
    {
        const int wave = tid >> 5;            // 2 waves per block
        const int lane = tid & 31;
        const bool hi  = (lane >> 4) != 0;    // false: K=0,1 ; true: K=2,3
        const int mrow = lane & 15;
        for (int t = wave; t < 12; t += 2) {  // 4 M-tiles x 3 N-tiles
            const int mt = t / 3, nt = t % 3;
            const float4 pj = s_pj4[mt*16 + mrow];   // ds_load_b128
            const float4 pk = s_pk4[nt*16 + mrow];
            v2f af, bf;
            af.x = hi ? pj.z : pj.x;
            af.y = hi ? 0.0f : pj.y;
            bf.x = hi ? pk.z : pk.x;
            bf.y = hi ? 0.0f : pk.y;
            v8f cacc = {};
            cacc = __builtin_amdgcn_wmma_f32_16x16x4_f32(
                false, af, false, bf, (short)0, cacc, false, false);
            const int    ncol = nt*16 + (lane & 15);
            const float4 kd   = s_kd[ncol];
            #pragma unroll
            for (int v = 0; v < 8; ++v) {
                const int    mr = mt*16 + v + (hi ? 8 : 0);
                const float4 jd = s_jd[mr];
                const float  r2jk = fmaxf(jd.z + kd.z - 2.0f * cacc[v], EPSf);
                float cv = (jd.x + kd.x - r2jk) * jd.y * kd.y;  // /(2 rij rik)
                cv = fminf(1.0f, fmaxf(-1.0f, cv));
                s_cos[mr][ncol] = cv;                 // unconditional store
            }
        }
    }
    __syncthreads();

    // -------- phase C: 21 angular gaussians, scatter into 5 element chans --
    if (tid < NJ_) {
        const int j = tid;
        float acc[5 * NAP_];
        #pragma unroll
        for (int i = 0; i < 5*NAP_; ++i) acc[i] = 0.0f;

        const float fj = s_jd[j].w;
        for (int k = 0; k < NK_; ++k) {
            const float w  = fj * s_kd[k].w * s_pm[j*NK_ + k];
            const float cv = s_cos[j][k];
            const float z0 = s_zk[k][0], z1 = s_zk[k][1], z2 = s_zk[k][2];
            const float z3 = s_zk[k][3], z4 = s_zk[k][4];
            #pragma unroll
            for (int g = 0; g < NAP_; ++g) {
                const float df = cv - (0.1f * (float)g - 1.0f);   // linspace(-1,1,21)
                const float wg = w * __expf(-25.0f * df * df);    // coeff = -0.5/(2*0.1^2)
                acc[0*NAP_ + g] += z0 * wg;
                acc[1*NAP_ + g] += z1 * wg;
                acc[2*NAP_ + g] += z2 * wg;
                acc[3*NAP_ + g] += z3 * wg;
                acc[4*NAP_ + g] += z4 * wg;
            }
        }
        const long long base = (a < NA_)
            ? OFF_APA + ((long long)(b*NA_ + a)        * NJ_ + j) * (5*NAP_)
            : OFF_APB + ((long long)(b*NB_ + (a-NA_))  * NJ_ + j) * (5*NAP_);
        #pragma unroll
        for (int i = 0; i < 5*NAP_; ++i) out[base + i] = acc[i];
    }
}

// ============================ K3: dists =====================================
__global__ __launch_bounds__(64) void apnet_dists(
    const float* __restrict__ positions,
    const float* __restrict__ cell,
    const float* __restrict__ noi,              // neighbor_offset_inter (B,A,NJ,3)
    const int*   __restrict__ nbr_inter,        // (B,A,NJ)
    const float* __restrict__ nbr_inter_mask,   // (B,A,NJ)
    float*       __restrict__ out)
{
    const int bna = blockIdx.x;                 // b*NA + a, a < 60
    const int b   = bna / NA_;
    const int a   = bna % NA_;
    const int j   = threadIdx.x;
    if (j >= NJ_) return;

    const long long aba = (long long)(b*A_ + a);
    const float* cb = cell + b * 9;
    const float c00=cb[0],c01=cb[1],c02=cb[2],c10=cb[3],c11=cb[4],c12=cb[5],c20=cb[6],c21=cb[7],c22=cb[8];
    const float pix = positions[aba*3+0];
    const float piy = positions[aba*3+1];
    const float piz = positions[aba*3+2];

    const int    jj = nbr_inter[aba*NJ_ + j];
    const float* pp = positions + (long long)(b*A_ + jj) * 3;
    const float* of = noi + (aba*NJ_ + j) * 3;
    const float dx = pp[0] - pix + of[0]*c00 + of[1]*c10 + of[2]*c20;
    const float dy = pp[1] - piy + of[0]*c01 + of[1]*c11 + of[2]*c21;
    const float dz = pp[2] - piz + of[0]*c02 + of[1]*c12 + of[2]*c22;
    const float d  = __builtin_sqrtf(fmaxf(dx*dx + dy*dy + dz*dz, EPSf));
    const float m  = nbr_inter_mask[aba*NJ_ + j];
    const float pd = (m != 0.0f) ? d : 1.0f;
    const float fc = fcut(pd, PI_9, 9.0f);

    const long long base = OFF_D + ((long long)b*(NA_*NB_) + a*NB_ + j) * 2;
    out[base + 0] = pd * fc;
    out[base + 1] = fc / pd;
}

// ============================ launcher ======================================
extern "C" void kernel_launch(void* const* d_in, const int* in_sizes, int n_in,
                              void* d_out, int out_size, void* d_ws, size_t ws_size,
                              hipStream_t stream) {
    const float* positions       = (const float*)d_in[0];
    const float* cell            = (const float*)d_in[1];
    const float* cell_offset     = (const float*)d_in[2];
    const float* noi             = (const float*)d_in[3];   // neighbor_offset_inter
    const float* coi             = (const float*)d_in[4];   // cell_offset_intra
    const float* z_weight        = (const float*)d_in[5];
    const int*   Zat             = (const int*)  d_in[6];
    const int*   neighbors       = (const int*)  d_in[7];
    const float* nbr_mask        = (const float*)d_in[8];
    // d_in[9] = ZA, d_in[10] = ZB : unused in forward
    const int*   idx_j           = (const int*)  d_in[11];
    const int*   idx_k           = (const int*)  d_in[12];
    const float* pairs_mask      = (const float*)d_in[13];
    const int*   nbr_inter       = (const int*)  d_in[14];
    const float* nbr_inter_mask  = (const float*)d_in[15];
    float* out = (float*)d_out;

    apnet_radial<<<B_ * A_, 64, 0, stream>>>(
        positions, cell, cell_offset, z_weight, Zat, neighbors, nbr_mask, out);

    apnet_angular<<<B_ * A_, 64, 0, stream>>>(
        positions, cell, noi, coi, z_weight, Zat, idx_j, idx_k, pairs_mask, out);

    apnet_dists<<<B_ * NA_, 64, 0, stream>>>(
        positions, cell, noi, nbr_inter, nbr_inter_mask, out);
}